// SoftPatternClassifier_24412594110565
// MI455X (gfx1250) — compile-verified
//
#include <hip/hip_runtime.h>
#include <math.h>

typedef __attribute__((ext_vector_type(2))) float v2f;
typedef __attribute__((ext_vector_type(8))) float v8f;

#define S_LEN 4096
#define DIM   300
#define NPAT  2000
#define LM1   5      // L-1 window length
#define HID   100
#define NCLS  2
#define DPAD  304    // padded row stride for LDS D tiles

// ---------------------------------------------------------------------------
// Phase 0: gather X = embeddings[doc]  -> workspace (S_LEN x DIM, row major)
// ---------------------------------------------------------------------------
__global__ void gather_x(const int* __restrict__ doc,
                         const float* __restrict__ emb,
                         float* __restrict__ xg) {
    int idx = blockIdx.x * blockDim.x + threadIdx.x;
    if (idx >= S_LEN * DIM) return;
    int s = idx / DIM, dcol = idx - s * DIM;
    xg[idx] = emb[(long)doc[s] * DIM + dcol];
}

// ---------------------------------------------------------------------------
// Phase 1: fused GEMM + sigmoid + sliding-window product.
// One wave (32 lanes) per 16 patterns. grid.x = NPAT/16 = 125.
//   score[p] = sum_s prod_{l=0..4} sigmoid(x[s+l] . d[p,l] + b[p,l])
// GEMM tiles via V_WMMA_F32_16X16X4_F32 (native CDNA5 f32 matrix op).
// ---------------------------------------------------------------------------
__global__ void __launch_bounds__(32)
pattern_scan(const float* __restrict__ xg,
             const float* __restrict__ diags,
             const float* __restrict__ bias,
             float* __restrict__ scores) {
    // Pattern matrices resident in LDS for the whole S loop (~97 KB),
    // plus a 20-row sliding window buffer per l, plus a tiny reducer.
    __shared__ float sD[LM1][16][DPAD];
    __shared__ float sAw[LM1][20][16];
    __shared__ float sRed[32];

    const int lane = threadIdx.x & 31;
    const int col  = lane & 15;   // pattern within tile (B/C/D column), X row (A)
    const int half = lane >> 4;   // lane group: selects K pair (A/B), row half (C/D)
    const int p0   = blockIdx.x * 16;

    // Stage D tile: rows (p*3+1)*6 + l of diags (diag index 1, positions 0..4).
    for (int idx = lane; idx < LM1 * 16 * (DIM / 2); idx += 32) {
        int l   = idx / (16 * (DIM / 2));
        int r   = idx - l * (16 * (DIM / 2));
        int pr  = r / (DIM / 2);
        int kk  = (r - pr * (DIM / 2)) * 2;
        int row = ((p0 + pr) * 3 + 1) * 6 + l;
        v2f v = *(const v2f*)(diags + (long)row * DIM + kk);
        sD[l][pr][kk]     = v.x;
        sD[l][pr][kk + 1] = v.y;
    }

    // Per-lane bias for its pattern column.
    float bl[LM1];
#pragma unroll
    for (int l = 0; l < LM1; ++l)
        bl[l] = bias[((p0 + col) * 3 + 1) * 6 + l];

    const v8f vzero = {0.f, 0.f, 0.f, 0.f, 0.f, 0.f, 0.f, 0.f};
    float scoreAcc = 0.0f;

    for (int c = 0; c < S_LEN; c += 16) {
        v8f acc[LM1];
#pragma unroll
        for (int l = 0; l < LM1; ++l) acc[l] = vzero;

        // A-frag base: X row (c+col), K pair offset 2*half. 8B-aligned.
        const float* xrow = xg + (long)(c + col) * DIM + 2 * half;

        for (int k = 0; k < DIM; k += 4) {
            v2f a = *(const v2f*)(xrow + k);               // A: 16x4 f32 frag
#pragma unroll
            for (int l = 0; l < LM1; ++l) {
                v2f b = *(const v2f*)(&sD[l][col][k + 2 * half]);  // B: 4x16 frag
                acc[l] = __builtin_amdgcn_wmma_f32_16x16x4_f32(
                    /*neg_a=*/false, a, /*neg_b=*/false, b,
                    /*c_mod=*/(short)0, acc[l],
                    /*reuse_a=*/false, /*reuse_b=*/false);
            }
        }

        // Bias + sigmoid, scatter into ring buffer rows 4..19
        // (D layout: VGPR v -> row v + 8*half, column = col).
#pragma unroll
        for (int l = 0; l < LM1; ++l) {
#pragma unroll
            for (int v = 0; v < 8; ++v) {
                float t = acc[l][v] + bl[l];
                t = 1.0f / (1.0f + __expf(-t));
                sAw[l][4 + v + 8 * half][col] = t;
            }
        }

        // Window products: buffer row i holds global row c-4+i.
        // Lane handles pattern `col`, window starts srel = 8*half + 0..7.
#pragma unroll
        for (int i = 0; i < 8; ++i) {
            int srel   = 8 * half + i;
            int sstart = c - 4 + srel;
            if (sstart >= 0 && sstart <= S_LEN - LM1) {
                float prod = sAw[0][srel][col];
                prod *= sAw[1][srel + 1][col];
                prod *= sAw[2][srel + 2][col];
                prod *= sAw[3][srel + 3][col];
                prod *= sAw[4][srel + 4][col];
                scoreAcc += prod;
            }
        }

        // Carry last 4 rows (global c+12..c+15) to the front for next chunk.
        for (int idx = lane; idx < LM1 * 4 * 16; idx += 32) {
            int l  = idx >> 6;
            int r  = (idx >> 4) & 3;
            int cc = idx & 15;
            sAw[l][r][cc] = sAw[l][16 + r][cc];
        }
    }

    // Combine the two row-halves per pattern column and store.
    sRed[lane] = scoreAcc;
    if (lane < 16) scores[p0 + lane] = sRed[lane] + sRed[lane + 16];
}

// ---------------------------------------------------------------------------
// Phase 2: MLP head  (1,P) @ (P,H) -> relu -> (H,C). Single block.
// ---------------------------------------------------------------------------
__global__ void __launch_bounds__(128)
mlp_head(const float* __restrict__ scores,
         const float* __restrict__ w1, const float* __restrict__ b1,
         const float* __restrict__ w2, const float* __restrict__ b2,
         float* __restrict__ out) {
    __shared__ float hid[HID];
    int t = threadIdx.x;
    if (t < HID) {
        float a = b1[t];
        for (int p = 0; p < NPAT; ++p) a = fmaf(scores[p], w1[p * HID + t], a);
        hid[t] = a > 0.f ? a : 0.f;
    }
    __syncthreads();
    if (t < NCLS) {
        float o = b2[t];
        for (int h = 0; h < HID; ++h) o = fmaf(hid[h], w2[h * NCLS + t], o);
        out[t] = o;
    }
}

// ---------------------------------------------------------------------------
extern "C" void kernel_launch(void* const* d_in, const int* in_sizes, int n_in,
                              void* d_out, int out_size, void* d_ws, size_t ws_size,
                              hipStream_t stream) {
    const int*   doc   = (const int*)d_in[0];
    const float* emb   = (const float*)d_in[1];
    const float* diags = (const float*)d_in[2];
    const float* bias  = (const float*)d_in[3];
    const float* w1    = (const float*)d_in[4];
    const float* b1    = (const float*)d_in[5];
    const float* w2    = (const float*)d_in[6];
    const float* b2    = (const float*)d_in[7];

    float* xg     = (float*)d_ws;                                   // 4096*300 f32
    float* scores = (float*)((char*)d_ws +
                             (size_t)S_LEN * DIM * sizeof(float));  // 2000 f32

    gather_x<<<(S_LEN * DIM + 255) / 256, 256, 0, stream>>>(doc, emb, xg);
    pattern_scan<<<NPAT / 16, 32, 0, stream>>>(xg, diags, bias, scores);
    mlp_head<<<1, 128, 0, stream>>>(scores, w1, b1, w2, b2, (float*)d_out);
}